// SubjectLayers_5626407158064
// MI455X (gfx1250) — compile-verified
//
#include <hip/hip_runtime.h>

// Problem constants (match reference): B=256, C=270, D=270, S=128, T=1024
#define B_ 256
#define C_ 270
#define D_ 270
#define T_ 1024

#define KC 16    // K (=C) chunk staged in LDS per iteration
#define DB 64    // d-rows per workgroup macro-tile
#define TB 128   // t-cols per workgroup macro-tile
#define XPAD 4   // LDS row padding for x tile (row stride 132 floats = 528B, 16B aligned)

typedef __attribute__((ext_vector_type(2))) float v2f;
typedef __attribute__((ext_vector_type(4))) float v4f;
typedef __attribute__((ext_vector_type(8))) float v8f;

// Builtin parameter type per hipcc diagnostic: __vector_size__(4*sizeof(int)) int *
typedef int v4i_gcc __attribute__((vector_size(16)));
typedef __attribute__((address_space(1))) v4i_gcc* gas_v4i;  // global
typedef __attribute__((address_space(3))) v4i_gcc* las_v4i;  // LDS

#if defined(__gfx1250__) && __has_builtin(__builtin_amdgcn_global_load_async_to_lds_b128)
#define HAVE_ASYNC_LDS 1
#else
#define HAVE_ASYNC_LDS 0
#endif

__device__ __forceinline__ void wait_asynccnt0() {
#if HAVE_ASYNC_LDS
#if __has_builtin(__builtin_amdgcn_s_wait_asynccnt)
    __builtin_amdgcn_s_wait_asynccnt(0);
#else
    asm volatile("s_wait_asynccnt 0x0" ::: "memory");
#endif
#endif
}

// out[b, d, t] = sum_c x[b, c, t] * weights[subjects[b], c, d]
// Per block: one batch b, output tile [d0:d0+64) x [t0:t0+128).
// 8 waves: wave = wd (0..3, 16-row d-tile) + 4*wt (0..1, 64-col t-half).
__global__ __launch_bounds__(256) void subject_layers_wmma(
    const float* __restrict__ x,       // [B, C, T]
    const int*   __restrict__ subjects,// [B]
    const float* __restrict__ weights, // [S, C, D]
    float* __restrict__ out)           // [B, D, T]
{
    __shared__ __align__(16) float w_s[2][KC][DB];        // W chunk: [c][d] (A source)
    __shared__ __align__(16) float x_s[2][KC][TB + XPAD]; // X chunk: [c][t] (B source)

    const int t0 = blockIdx.x * TB;
    const int d0 = blockIdx.y * DB;
    const int b  = blockIdx.z;

    const int subj = subjects[b];
    const float* __restrict__ wsub = weights + (size_t)subj * (C_ * D_);
    const float* __restrict__ xb   = x       + (size_t)b    * (C_ * T_);
    float*       __restrict__ ob   = out     + (size_t)b    * (D_ * T_);

    const int tid  = threadIdx.x;
    const int lane = tid & 31;
    const int wave = tid >> 5;
    const int wd   = wave & 3;   // 16-row d-tile inside the 64-row block
    const int wt   = wave >> 2;  // 64-col t-half inside the 128-col block

    const int m  = lane & 15;    // row/col index within a 16-wide fragment
    const int kh = lane >> 4;    // K-half select (ISA A/B fragment layout)

    // Staging thread mapping (same every chunk)
    const int wrow = tid >> 4;          // 0..15  (W: c-row)
    const int wcol = (tid & 15) << 2;   // 0..60  (W: 4 d-cols)
    const int xrow = tid >> 5;          // 0..7   (X: c-row, two halves)
    const int xcol = (tid & 31) << 2;   // 0..124 (X: 4 t-cols)

    v8f acc[4];
#pragma unroll
    for (int i = 0; i < 4; ++i) acc[i] = (v8f){0.f,0.f,0.f,0.f,0.f,0.f,0.f,0.f};

    const int nchunks = (C_ + KC - 1) / KC; // 17 (last chunk partial)

    // ---------- staging helpers (inline) ----------
    auto stage_w = [&](int c0, int buf) {
        const int c = c0 + wrow;
        v4f wv = (v4f){0.f, 0.f, 0.f, 0.f};
        if (c < C_) {
            if (d0 + wcol + 3 < D_) {
                wv = *(const v4f*)&wsub[c * D_ + d0 + wcol];   // interior fast path
            } else {
#pragma unroll
                for (int j = 0; j < 4; ++j) {
                    const int d = d0 + wcol + j;
                    wv[j] = (d < D_) ? wsub[c * D_ + d] : 0.f; // exact zero-fill at edges
                }
            }
        }
        *(v4f*)&w_s[buf][wrow][wcol] = wv;
    };

    auto stage_x = [&](int c0, int buf) {
#pragma unroll
        for (int half = 0; half < 2; ++half) {
            const int c    = c0 + xrow + half * 8;
            const int crow = (c < C_) ? c : 0; // clamp: matching A rows are zero, garbage*0=0
#if HAVE_ASYNC_LDS
            __builtin_amdgcn_global_load_async_to_lds_b128(
                (gas_v4i)&xb[(size_t)crow * T_ + t0 + xcol],
                (las_v4i)&x_s[buf][xrow + half * 8][xcol],
                0, 0);
#else
            *(v4f*)&x_s[buf][xrow + half * 8][xcol] =
                *(const v4f*)&xb[(size_t)crow * T_ + t0 + xcol];
#endif
        }
    };
    // ----------------------------------------------

    // Prologue: fill buffer 0 with chunk 0
    stage_w(0, 0);
    stage_x(0, 0);
    wait_asynccnt0();
    __syncthreads();

    for (int ck = 0; ck < nchunks; ++ck) {
        const int cur = ck & 1;

        // Software pipeline: kick off chunk ck+1 into the other buffer
        if (ck + 1 < nchunks) {
            stage_w((ck + 1) * KC, cur ^ 1);
            stage_x((ck + 1) * KC, cur ^ 1);
        }

        // ---- compute chunk ck: 4 k-steps x 4 WMMAs (A reused per step) ----
#pragma unroll
        for (int kk = 0; kk < KC; kk += 4) {
            const int k0 = kk + kh * 2; // lanes 0-15: K=kk+0/1, lanes 16-31: K=kk+2/3
            v2f a;
            a[0] = w_s[cur][k0 + 0][wd * 16 + m];
            a[1] = w_s[cur][k0 + 1][wd * 16 + m];
#pragma unroll
            for (int tt = 0; tt < 4; ++tt) {
                v2f bm;
                bm[0] = x_s[cur][k0 + 0][wt * 64 + tt * 16 + m];
                bm[1] = x_s[cur][k0 + 1][wt * 64 + tt * 16 + m];
                // 8 args: (neg_a, A, neg_b, B, c_mod, C, reuse_a, reuse_b)
                acc[tt] = __builtin_amdgcn_wmma_f32_16x16x4_f32(
                    false, a, false, bm, (short)0, acc[tt], false, false);
            }
        }

        // Next chunk's async copies must have landed before anyone reads them
        wait_asynccnt0();
        __syncthreads();
    }

    // ---- store: wave tile rows d0+wd*16+{r+8*kh}, cols t0+wt*64+tt*16+m ----
#pragma unroll
    for (int tt = 0; tt < 4; ++tt) {
#pragma unroll
        for (int r = 0; r < 8; ++r) {
            const int d = d0 + wd * 16 + r + 8 * kh;
            const int t = t0 + wt * 64 + tt * 16 + m;
            if (d < D_) ob[(size_t)d * T_ + t] = acc[tt][r];
        }
    }
}

extern "C" void kernel_launch(void* const* d_in, const int* in_sizes, int n_in,
                              void* d_out, int out_size, void* d_ws, size_t ws_size,
                              hipStream_t stream) {
    const float* x        = (const float*)d_in[0]; // [B, C, T] fp32
    const int*   subjects = (const int*)  d_in[1]; // [B] int
    const float* weights  = (const float*)d_in[2]; // [S, C, D] fp32
    float*       out      = (float*)d_out;         // [B, D, T] fp32

    dim3 grid(T_ / TB, (D_ + DB - 1) / DB, B_); // (8, 5, 256)
    subject_layers_wmma<<<grid, 256, 0, stream>>>(x, subjects, weights, out);
}